// ProbAttention_1657857376403
// MI455X (gfx1250) — compile-verified
//
#include <hip/hip_runtime.h>
#include <math.h>

typedef __attribute__((ext_vector_type(2))) float v2f;
typedef __attribute__((ext_vector_type(8))) float v8f;

#define B_ 4
#define L_ 4096
#define H_ 8
#define D_ 64
#define NPAD 48
#define SCALE 0.125f  /* 1/sqrt(64) */

__device__ __forceinline__ float wave_reduce_add32(float v) {
  for (int off = 16; off > 0; off >>= 1) v += __shfl_xor(v, off, 32);
  return v;
}

// ---------------------------------------------------------------------------
// V.mean over L per (b,h,d)  ->  meanV[B*H, D]
// ---------------------------------------------------------------------------
__global__ void mean_v_kernel(const float* __restrict__ V, float* __restrict__ meanV) {
  int bh = blockIdx.x;
  int b = bh / H_, h = bh % H_;
  int t = threadIdx.x;       // 256
  int d = t & 63;
  int chunk = t >> 6;        // 4 chunks of L/4
  float s = 0.f;
  int l0 = chunk * (L_ / 4);
  for (int l = l0; l < l0 + L_ / 4; ++l)
    s += V[((size_t)(b * L_ + l) * H_ + h) * D_ + d];
  __shared__ float red[256];
  red[t] = s;
  __syncthreads();
  if (chunk == 0) {
    float tot = red[d] + red[d + 64] + red[d + 128] + red[d + 192];
    meanV[bh * D_ + d] = tot * (1.0f / L_);
  }
}

// ---------------------------------------------------------------------------
// context[b,h,l,d] = meanV[b,h,d]   (B,H,L,D layout)
// ---------------------------------------------------------------------------
__global__ void fill_ctx_kernel(const float* __restrict__ meanV, float* __restrict__ out) {
  size_t i = (size_t)blockIdx.x * blockDim.x + threadIdx.x;
  int d = (int)(i & 63);
  int bh = (int)(i >> 18);   // / (L*D) = 2^18
  out[i] = meanV[bh * D_ + d];
}

// ---------------------------------------------------------------------------
// M[b,h,l] = max_s(q.k_s) - mean over L of sampled dots.  One wave per query.
// ---------------------------------------------------------------------------
__global__ void sparsity_m_kernel(const float* __restrict__ Q, const float* __restrict__ K,
                                  const int* __restrict__ idx, float* __restrict__ M, int S) {
  int wave = threadIdx.x >> 5;
  int lane = threadIdx.x & 31;
  int g = blockIdx.x * (blockDim.x >> 5) + wave;   // [0, B*H*L)
  int bh = g >> 12;          // / L
  int l  = g & (L_ - 1);
  int b = bh / H_, h = bh % H_;
  const float2* qrow = (const float2*)(Q + ((size_t)(b * L_ + l) * H_ + h) * D_);
  float2 q = qrow[lane];
  float mx = -INFINITY, sm = 0.f;
  const int* irow = idx + (size_t)l * S;
  for (int s = 0; s < S; ++s) {
    int j = irow[s];
    const float2* krow = (const float2*)(K + ((size_t)(b * L_ + j) * H_ + h) * D_);
    float2 k = krow[lane];
    float p = wave_reduce_add32(q.x * k.x + q.y * k.y);
    mx = fmaxf(mx, p);
    sm += p;
  }
  if (lane == 0) M[(size_t)bh * L_ + l] = mx - sm * (1.0f / L_);
}

// ---------------------------------------------------------------------------
// Exact top-k (iterative argmax, ties -> smallest index).  One WG per (b,h).
// Destroys M (scratch; recomputed every call).
// ---------------------------------------------------------------------------
__global__ void topk_kernel(float* __restrict__ M, int* __restrict__ Mtop, int ntop) {
  int bh = blockIdx.x;
  int tid = threadIdx.x;     // 256
  float* Mrow = M + (size_t)bh * L_;
  __shared__ float sv[256];
  __shared__ int   si[256];
  for (int t = 0; t < ntop; ++t) {
    float bv = -INFINITY; int bi = 0x7fffffff;
    for (int i = tid; i < L_; i += 256) {
      float v = Mrow[i];
      if (v > bv) { bv = v; bi = i; }   // strict > keeps smallest index on tie
    }
    sv[tid] = bv; si[tid] = bi;
    __syncthreads();
    for (int s = 128; s > 0; s >>= 1) {
      if (tid < s) {
        float ov = sv[tid + s]; int oi = si[tid + s];
        if (ov > sv[tid] || (ov == sv[tid] && oi < si[tid])) { sv[tid] = ov; si[tid] = oi; }
      }
      __syncthreads();
    }
    if (tid == 0) { Mtop[bh * NPAD + t] = si[0]; Mrow[si[0]] = -INFINITY; }
    __syncthreads();
  }
  if (tid == 0)
    for (int t = ntop; t < NPAD; ++t) Mtop[bh * NPAD + t] = -1;
}

// ---------------------------------------------------------------------------
// Flash-style attention for the 45 (padded 48) selected queries per (b,h).
// WG = 128 threads (4 waves): waves 0..2 each own a 16-row m-tile; all waves
// cooperatively stage K/V 16x64 tiles in LDS.  fp32 WMMA 16x16x4 throughout.
// ---------------------------------------------------------------------------
__global__ __launch_bounds__(128)
void prob_attn_kernel(const float* __restrict__ Q, const float* __restrict__ K,
                      const float* __restrict__ V, const int* __restrict__ Mtop,
                      float* __restrict__ out, int ntop) {
  int bh = blockIdx.x;
  int b = bh / H_, h = bh % H_;
  int tid  = threadIdx.x;
  int wave = tid >> 5;
  int lane = tid & 31;
  int half = lane >> 4;      // K-pair select in A/B fragments
  int lcol = lane & 15;

  __shared__ float Kt[16 * 68];      // padded stride 68 -> bank-conflict free
  __shared__ float Vt[16 * 68];
  __shared__ float Pt[3][16 * 17];   // per-wave P round-trip, stride 17
  __shared__ int   topIdx[NPAD];

  if (tid < NPAD) topIdx[tid] = Mtop[bh * NPAD + tid];
  __syncthreads();

  // Resident A-fragments of Q_reduce (scaled): qa[g] covers K-cols 4g..4g+3.
  // ISA 16x4 f32 A layout: lane -> M = lane%16, half-wave -> K pair.
  v2f qa[16];
  int myRow = 16 * wave + lcol;
  if (wave < 3) {
    int qidx = (myRow < ntop) ? topIdx[myRow] : -1;
    if (qidx >= 0) {
      const float2* qrow = (const float2*)(Q + ((size_t)(b * L_ + qidx) * H_ + h) * D_);
      #pragma unroll
      for (int g = 0; g < 16; ++g) {
        float2 qv = qrow[2 * g + half];
        qa[g].x = qv.x * SCALE; qa[g].y = qv.y * SCALE;
      }
    } else {
      #pragma unroll
      for (int g = 0; g < 16; ++g) { qa[g].x = 0.f; qa[g].y = 0.f; }
    }
  }

  v8f acc[4] = {};                  // 16 x 64 output accumulator (4 N-tiles)
  float mrow[8], lrow[8];           // per-row running max / sumexp
  #pragma unroll
  for (int i = 0; i < 8; ++i) { mrow[i] = -INFINITY; lrow[i] = 0.f; }

  for (int kt = 0; kt < L_ / 16; ++kt) {
    __syncthreads();                // previous tile fully consumed
    // Cooperative K/V tile load: 16 rows x 64 floats, float4 per thread x2.
    for (int q = tid; q < 256; q += 128) {
      int row = q >> 4, c4 = q & 15;
      size_t gbase = ((size_t)(b * L_ + kt * 16 + row) * H_ + h) * D_ + c4 * 4;
      float4 kv = *(const float4*)(K + gbase);
      float4 vv = *(const float4*)(V + gbase);
      float* kd = &Kt[row * 68 + c4 * 4];
      kd[0] = kv.x; kd[1] = kv.y; kd[2] = kv.z; kd[3] = kv.w;
      float* vd = &Vt[row * 68 + c4 * 4];
      vd[0] = vv.x; vd[1] = vv.y; vd[2] = vv.z; vd[3] = vv.w;
    }
    __syncthreads();

    if (wave < 3) {
      // ---- S = (Q*scale) * K^T : 16 chained v_wmma_f32_16x16x4_f32 ----
      v8f s = {};
      #pragma unroll
      for (int g = 0; g < 16; ++g) {
        int kk = 4 * g + 2 * half;
        v2f kb;                                   // B[kk][n] = K[n][kk]
        kb.x = Kt[lcol * 68 + kk];
        kb.y = Kt[lcol * 68 + kk + 1];
        s = __builtin_amdgcn_wmma_f32_16x16x4_f32(false, qa[g], false, kb,
                                                  (short)0, s, false, false);
      }
      // ---- online softmax: C-tile row r lives in VGPR r%8, half-wave r/8 ----
      float corr[8];
      #pragma unroll
      for (int i = 0; i < 8; ++i) {
        float rmax = s[i];
        for (int off = 8; off > 0; off >>= 1) rmax = fmaxf(rmax, __shfl_xor(rmax, off, 16));
        float mnew = fmaxf(mrow[i], rmax);
        float c = __expf(mrow[i] - mnew);
        float p = __expf(s[i] - mnew);
        float rs = p;
        for (int off = 8; off > 0; off >>= 1) rs += __shfl_xor(rs, off, 16);
        lrow[i] = lrow[i] * c + rs;
        mrow[i] = mnew;
        corr[i] = c;
        Pt[wave][(i + 8 * half) * 17 + lcol] = p;   // P in row-major LDS
      }
      #pragma unroll
      for (int i = 0; i < 8; ++i) {
        acc[0][i] *= corr[i]; acc[1][i] *= corr[i];
        acc[2][i] *= corr[i]; acc[3][i] *= corr[i];
      }
      // Wave-local DS RAW: DS ops are in-order per wave; wait + compiler fence.
      __asm__ volatile("s_wait_dscnt 0" ::: "memory");
      // ---- acc += P(16x16) * V(16x64) : 4 K-groups x 4 N-tiles ----
      #pragma unroll
      for (int g = 0; g < 4; ++g) {
        int kk = 4 * g + 2 * half;
        v2f pa;                                   // A[m][kk] = P[m][kk]
        pa.x = Pt[wave][lcol * 17 + kk];
        pa.y = Pt[wave][lcol * 17 + kk + 1];
        #pragma unroll
        for (int nt = 0; nt < 4; ++nt) {
          v2f vb;                                 // B[kk][n] = V[kk][16*nt+n]
          vb.x = Vt[kk * 68 + 16 * nt + lcol];
          vb.y = Vt[(kk + 1) * 68 + 16 * nt + lcol];
          acc[nt] = __builtin_amdgcn_wmma_f32_16x16x4_f32(false, pa, false, vb,
                                                          (short)0, acc[nt], false, false);
        }
      }
    }
  }

  // Epilogue: normalize and scatter the valid rows into context.
  if (wave < 3) {
    #pragma unroll
    for (int i = 0; i < 8; ++i) {
      int row = 16 * wave + i + 8 * half;
      if (row < ntop) {
        int qidx = topIdx[row];
        float inv = 1.0f / lrow[i];
        size_t obase = ((size_t)(b * H_ + h) * L_ + qidx) * D_;
        out[obase + 0 * 16 + lcol] = acc[0][i] * inv;
        out[obase + 1 * 16 + lcol] = acc[1][i] * inv;
        out[obase + 2 * 16 + lcol] = acc[2][i] * inv;
        out[obase + 3 * 16 + lcol] = acc[3][i] * inv;
      }
    }
  }
}

// ---------------------------------------------------------------------------
extern "C" void kernel_launch(void* const* d_in, const int* in_sizes, int n_in,
                              void* d_out, int out_size, void* d_ws, size_t ws_size,
                              hipStream_t stream) {
  (void)n_in; (void)out_size; (void)ws_size;
  const float* Q  = (const float*)d_in[0];
  const float* K  = (const float*)d_in[1];
  const float* V  = (const float*)d_in[2];
  const int* idx  = (const int*)d_in[3];
  float* out      = (float*)d_out;

  int S    = in_sizes[3] / L_;   // sample_k (= 45)
  int ntop = S;                  // n_top uses the same formula (= 45, <= NPAD)

  float* meanV = (float*)d_ws;                         // B*H*D floats
  float* M     = meanV + B_ * H_ * D_;                 // B*H*L floats
  int*   Mtop  = (int*)(M + (size_t)B_ * H_ * L_);     // B*H*NPAD ints

  mean_v_kernel<<<B_ * H_, 256, 0, stream>>>(V, meanV);
  fill_ctx_kernel<<<(B_ * H_ * L_ * D_) / 256, 256, 0, stream>>>(meanV, out);
  sparsity_m_kernel<<<(B_ * H_ * L_) / 8, 256, 0, stream>>>(Q, K, idx, M, S);
  topk_kernel<<<B_ * H_, 256, 0, stream>>>(M, Mtop, ntop);
  prob_attn_kernel<<<B_ * H_, 128, 0, stream>>>(Q, K, V, Mtop, out, ntop);
}